// RoiMaxPooling_88965952569946
// MI455X (gfx1250) — compile-verified
//
#include <hip/hip_runtime.h>

// RoiMaxPooling: (256,56,56,256) fp32 -> 8x8 adaptive max pool -> 7x7 bins,
// with the reference's output-row permutation n_out = (w*7 + h)*256 + b.
//
// Pure streaming reduction (822 MB in / 12.8 MB out, ~36 us at 23.3 TB/s).
// CDNA5 async memory path: global_load_async_to_lds_b128 with triple-buffered
// ASYNCcnt pipelining; each wave reduces its own LDS slice (no barriers).
// s_wait_dscnt 0 before each recycled-buffer issue closes the WAR window
// between prior ds_loads and the async engine's LDS writes.

#define N_BATCH 256
#define HW      56
#define NCH     256

__device__ __forceinline__ void async_load_b128(unsigned lds_addr, const void* gaddr) {
    // VDST = LDS byte address (AS3 offset), VADDR = 64-bit global address, saddr = off
    asm volatile("global_load_async_to_lds_b128 %0, %1, off"
                 :: "v"(lds_addr), "v"(gaddr)
                 : "memory");
}
__device__ __forceinline__ void wait_async_0() { asm volatile("s_wait_asynccnt 0" ::: "memory"); }
__device__ __forceinline__ void wait_async_4() { asm volatile("s_wait_asynccnt 4" ::: "memory"); }
__device__ __forceinline__ void wait_async_8() { asm volatile("s_wait_asynccnt 8" ::: "memory"); }
__device__ __forceinline__ void wait_ds_0()    { asm volatile("s_wait_dscnt 0"    ::: "memory"); }

__global__ __launch_bounds__(256)
void roi_maxpool_async_kernel(const float* __restrict__ x, float* __restrict__ out) {
    // lds[wave][3-deep pingpong][pixel_in_tile][lane] : 8*3*4*32*16B = 48 KB
    __shared__ float4 lds[8][3][4][32];

    const int lane = threadIdx.x & 31;
    const int wv   = threadIdx.x >> 5;            // 0..7, uniform per wave
    const int half = wv & 1;                      // channel half: 0 -> ch 0..127, 1 -> 128..255

    // Bin coordinates: all scalar except the cheap per-wave b offset.
    const int w = blockIdx.x;                     // 0..6
    const int h = blockIdx.y;                     // 0..6
    const int b = blockIdx.z * 4 + (wv >> 1);     // 0..255

    // Byte address of x[b, h*8, w*8, half*128 + lane*4]
    const char* xb = (const char*)x
        + (size_t)((b * HW + h * 8) * HW + w * 8) * (NCH * 4)
        + half * 512 + lane * 16;

    float4 acc = make_float4(-INFINITY, -INFINITY, -INFINITY, -INFINITY);

    // pixel p = t*4 + j ; row offset = p>>3 (0..7), col offset = p&7 (0..7)
    #define ISSUE_TILE(t, buf)                                                 \
        _Pragma("unroll")                                                      \
        for (int j = 0; j < 4; ++j) {                                          \
            const int p  = (t) * 4 + j;                                        \
            const int ro = p >> 3, co = p & 7;                                 \
            unsigned laddr = (unsigned)(uintptr_t)&lds[wv][buf][j][lane];      \
            async_load_b128(laddr, xb + (size_t)ro * (HW * NCH * 4)            \
                                      + (size_t)co * (NCH * 4));               \
        }

    #define REDUCE_TILE(buf)                                                   \
        _Pragma("unroll")                                                      \
        for (int j = 0; j < 4; ++j) {                                          \
            float4 v = lds[wv][buf][j][lane];                                  \
            acc.x = fmaxf(acc.x, v.x);                                         \
            acc.y = fmaxf(acc.y, v.y);                                         \
            acc.z = fmaxf(acc.z, v.z);                                         \
            acc.w = fmaxf(acc.w, v.w);                                         \
        }

    // Triple-buffered pipeline over 16 tiles (4 pixels = 4 async loads each).
    ISSUE_TILE(0, 0)
    ISSUE_TILE(1, 1)
    #pragma unroll
    for (int t = 2; t < 16; ++t) {
        wait_ds_0();               // prior reduce's LDS reads done before buffer reuse
        ISSUE_TILE(t, t % 3)
        wait_async_8();            // in-order completion: tile t-2 has landed
        REDUCE_TILE((t - 2) % 3)
    }
    wait_async_4();                // tile 14 landed
    REDUCE_TILE(14 % 3)
    wait_async_0();                // tile 15 landed
    REDUCE_TILE(15 % 3)

    // Output row permutation: n = (w*7 + h)*256 + b ; float4 index within row.
    float4* o = (float4*)out;
    o[(size_t)((w * 7 + h) * N_BATCH + b) * (NCH / 4) + half * 32 + lane] = acc;

    #undef ISSUE_TILE
    #undef REDUCE_TILE
}

extern "C" void kernel_launch(void* const* d_in, const int* in_sizes, int n_in,
                              void* d_out, int out_size, void* d_ws, size_t ws_size,
                              hipStream_t stream) {
    (void)in_sizes; (void)n_in; (void)d_ws; (void)ws_size; (void)out_size;
    const float* x  = (const float*)d_in[0];
    float* out      = (float*)d_out;
    // grid: (w, h, b/4); block: 256 threads = 8 waves = 4 bins (2 waves per bin)
    dim3 grid(7, 7, N_BATCH / 4), block(256);
    roi_maxpool_async_kernel<<<grid, block, 0, stream>>>(x, out);
}